// BernoulliEdge_60936995995884
// MI455X (gfx1250) — compile-verified
//
#include <hip/hip_runtime.h>

typedef __attribute__((ext_vector_type(2))) float v2f;
typedef __attribute__((ext_vector_type(4))) float v4f;
typedef __attribute__((ext_vector_type(8))) float v8f;

#define B_ 16
#define N_ 1024
#define D_ 128

// Kernel 1: v[b][c] = b1[c] + sum_k nodes[b][n][k] * W1[k][c]   (left-half of concat, rank-1)
__global__ void be_vvec(const float* __restrict__ nodes, const float* __restrict__ W1,
                        const float* __restrict__ b1, const int* __restrict__ num_nodes,
                        float* __restrict__ vbuf) {
  int b = blockIdx.x;
  int c = threadIdx.x;                       // 0..127
  int n = num_nodes[b];
  const float* nd = nodes + ((size_t)b * N_ + n) * D_;
  float acc = b1[c];
  #pragma unroll 8
  for (int k = 0; k < D_; ++k) acc = fmaf(nd[k], W1[k * D_ + c], acc);
  vbuf[b * D_ + c] = acc;
}

// Kernel 2: per-batch GEMM  h = relu(nodes_b @ W1[128:,:] + v),  logits = h @ W2 + b2
// One wave per 16-row M-tile; WMMA f32 16x16x4 chained over K=128 (32 steps).
__global__ void be_gemm_logits(const float* __restrict__ nodes, const float* __restrict__ W1,
                               const float* __restrict__ W2, const float* __restrict__ b2,
                               const float* __restrict__ vbuf, float* __restrict__ logits) {
  const int lane  = threadIdx.x & 31;
  const int wave  = threadIdx.x >> 5;
  const int b     = blockIdx.y;
  const int mTile = blockIdx.x * 4 + wave;   // 0..63
  const int mBase = mTile * 16;
  const int half  = lane >> 4;               // 0: K pair {0,1}; 1: K pair {2,3}
  const int lm    = lane & 15;               // A: row within tile, B: column within tile

  // Preload A fragments for all 32 K-steps (held in VGPRs, reused across 8 N-tiles).
  const float* Arow = nodes + ((size_t)b * N_ + (mBase + lm)) * D_;
  v2f af[32];
  #pragma unroll
  for (int ks = 0; ks < 32; ++ks) {
    int kk = 4 * ks + 2 * half;
    af[ks] = *(const v2f*)(Arow + kk);
  }

  float lpa[8], lpb[8];
  #pragma unroll
  for (int r = 0; r < 8; ++r) { lpa[r] = 0.0f; lpb[r] = 0.0f; }

  const float* W1b = W1 + 128 * D_;          // bottom half of W1 (rows 128..255)

  for (int nT = 0; nT < 8; ++nT) {
    const int col = nT * 16 + lm;
    v8f acc = {};
    #pragma unroll
    for (int ks = 0; ks < 32; ++ks) {
      int kk = 4 * ks + 2 * half;
      v2f bv;
      bv.x = W1b[kk * D_ + col];
      bv.y = W1b[(kk + 1) * D_ + col];
      acc = __builtin_amdgcn_wmma_f32_16x16x4_f32(
          /*neg_a=*/false, af[ks], /*neg_b=*/false, bv,
          /*c_mod=*/(short)0, acc, /*reuse_a=*/false, /*reuse_b=*/false);
    }
    // h = relu(acc + v[col]); fold the @W2 contraction (per-column weights) into row partials.
    const float vc  = vbuf[b * D_ + col];
    const float w20 = W2[col * 2 + 0];
    const float w21 = W2[col * 2 + 1];
    #pragma unroll
    for (int r = 0; r < 8; ++r) {
      float h = fmaxf(acc[r] + vc, 0.0f);
      lpa[r] = fmaf(h, w20, lpa[r]);
      lpb[r] = fmaf(h, w21, lpb[r]);
    }
  }

  // Reduce across the 16 lanes of each half (xor masks < 16 stay within the half).
  #pragma unroll
  for (int r = 0; r < 8; ++r) {
    #pragma unroll
    for (int m = 8; m >= 1; m >>= 1) {
      lpa[r] += __shfl_xor(lpa[r], m, 32);
      lpb[r] += __shfl_xor(lpb[r], m, 32);
    }
  }
  if (lm == 0) {
    const float bb0 = b2[0], bb1 = b2[1];
    #pragma unroll
    for (int r = 0; r < 8; ++r) {
      int orow = mBase + r + 8 * half;       // C/D layout: lanes>=16 hold rows M=r+8
      size_t o = ((size_t)b * N_ + orow) * 2;
      logits[o + 0] = lpa[r] + bb0;
      logits[o + 1] = lpb[r] + bb1;
    }
  }
}

__device__ __forceinline__ v2f st_pair(float z0, float z1) {
  // softmax over 2 (max-subtracted, matching jax.nn.softmax), first-index argmax,
  // straight-through: y + (hard - y) with the reference's rounding order.
  float m  = fmaxf(z0, z1);
  float e0 = expf(z0 - m), e1 = expf(z1 - m);
  float sm = e0 + e1;
  float y0 = e0 / sm, y1 = e1 / sm;
  float h0 = (y0 >= y1) ? 1.0f : 0.0f;
  float h1 = 1.0f - h0;
  v2f p;
  p.x = y0 + (h0 - y0);
  p.y = y1 + (h1 - y1);
  return p;
}

// Kernel 3: scatter select + gumbel-softmax(hard) over B*N*N.
// Each thread handles two adjacent j positions (4 floats) -> b128 NT loads/stores.
__global__ void be_elementwise(const float* __restrict__ state, const float* __restrict__ gumbel,
                               const int* __restrict__ num_nodes, const float* __restrict__ logits,
                               float* __restrict__ outS, float* __restrict__ outP) {
  size_t t   = (size_t)blockIdx.x * blockDim.x + threadIdx.x;   // over B*N*N/2
  size_t idx = t * 2;                                           // (b,i,j) element pair base
  int b   = (int)(idx >> 20);
  int rem = (int)(idx & ((1u << 20) - 1));
  int i   = rem >> 10;
  int j0  = rem & 1023;                                         // even; j1 = j0+1
  int n   = num_nodes[b];

  v4f s = __builtin_nontemporal_load((const v4f*)state + t);
  const v2f* lg = (const v2f*)logits + (size_t)b * N_;
  // col mask: j == n && i <= n
  if (i <= n) {
    if (j0 == n)     { v2f l = lg[i]; s.x = l.x; s.y = l.y; }
    if (j0 + 1 == n) { v2f l = lg[i]; s.z = l.x; s.w = l.y; }
  }
  // row mask: i == n && j <= n  (applied last, overrides col at (n,n))
  if (i == n) {
    if (j0 <= n)     { v2f l = lg[j0];     s.x = l.x; s.y = l.y; }
    if (j0 + 1 <= n) { v2f l = lg[j0 + 1]; s.z = l.x; s.w = l.y; }
  }

  v4f g = __builtin_nontemporal_load((const v4f*)gumbel + t);
  v2f pa = st_pair(s.x + g.x, s.y + g.y);
  v2f pb = st_pair(s.z + g.z, s.w + g.w);
  v4f p; p.x = pa.x; p.y = pa.y; p.z = pb.x; p.w = pb.y;

  __builtin_nontemporal_store(s, (v4f*)outS + t);
  __builtin_nontemporal_store(p, (v4f*)outP + t);
}

extern "C" void kernel_launch(void* const* d_in, const int* in_sizes, int n_in,
                              void* d_out, int out_size, void* d_ws, size_t ws_size,
                              hipStream_t stream) {
  const float* nodes     = (const float*)d_in[0];
  const float* state     = (const float*)d_in[1];
  const float* W1        = (const float*)d_in[2];
  const float* b1        = (const float*)d_in[3];
  const float* W2        = (const float*)d_in[4];
  const float* b2        = (const float*)d_in[5];
  const int*   num_nodes = (const int*)  d_in[6];
  const float* gumbel    = (const float*)d_in[7];

  float* outS = (float*)d_out;
  float* outP = outS + (size_t)B_ * N_ * N_ * 2;

  float* vbuf   = (float*)d_ws;            // B*128 floats   (8 KB)
  float* logits = vbuf + B_ * D_;          // B*N*2 floats   (128 KB)

  hipLaunchKernelGGL(be_vvec, dim3(B_), dim3(D_), 0, stream,
                     nodes, W1, b1, num_nodes, vbuf);
  hipLaunchKernelGGL(be_gemm_logits, dim3(16, B_), dim3(128), 0, stream,
                     nodes, W1, W2, b2, vbuf, logits);
  size_t pairs = (size_t)B_ * N_ * N_ / 2; // 8,388,608 threads, 2 elements each
  hipLaunchKernelGGL(be_elementwise, dim3((unsigned)(pairs / 256)), dim3(256), 0, stream,
                     state, gumbel, num_nodes, logits, outS, outP);
}